// GCN_58823872086172
// MI455X (gfx1250) — compile-verified
//
#include <hip/hip_runtime.h>

#define HDIM 128  // hidden units (fixed by reference)

typedef __attribute__((ext_vector_type(2))) float v2f;
typedef __attribute__((ext_vector_type(8))) float v8f;

// ---------------- utility: zero fill ----------------
__global__ void gcn_fill_zero(float* __restrict__ p, int n) {
  int i = blockIdx.x * blockDim.x + threadIdx.x;
  if (i < n) p[i] = 0.0f;
}

// ---------------- degrees via atomic counts ----------------
__global__ void gcn_degrees(const int* __restrict__ src, const int* __restrict__ dst,
                            float* __restrict__ degO, float* __restrict__ degI, int E) {
  int i = blockIdx.x * blockDim.x + threadIdx.x;
  if (i < E) {
    atomicAdd(&degO[src[i]], 1.0f);
    atomicAdd(&degI[dst[i]], 1.0f);
  }
}

// inv = clip(deg,1,inf)^-0.5, in place on both arrays
__global__ void gcn_inv_sqrt(float* __restrict__ degO, float* __restrict__ degI, int n) {
  int i = blockIdx.x * blockDim.x + threadIdx.x;
  if (i < n) {
    degO[i] = 1.0f / sqrtf(fmaxf(degO[i], 1.0f));
    degI[i] = 1.0f / sqrtf(fmaxf(degI[i], 1.0f));
  }
}

// ---------------- embedding gather: x = z_table[z] ----------------
__global__ void gcn_gather_embed(const int* __restrict__ z, const float* __restrict__ table,
                                 float* __restrict__ x, int total) {
  int i = blockIdx.x * blockDim.x + threadIdx.x;
  if (i < total) {
    int node = i >> 7;          // / HDIM
    int t = i & (HDIM - 1);
    x[i] = table[(size_t)z[node] * HDIM + t];
  }
}

// ---------------- edge scatter: m[dst] += x[src] * inv_out[src] ----------------
// one wave (32 lanes) per edge; each lane moves a float4 (128 floats total)
__global__ void gcn_edge_scatter(const int* __restrict__ src, const int* __restrict__ dst,
                                 const float* __restrict__ x, const float* __restrict__ invO,
                                 float* __restrict__ m, int E) {
  int wid = (blockIdx.x * blockDim.x + threadIdx.x) >> 5;
  int lane = threadIdx.x & 31;
  if (wid >= E) return;
  int s = src[wid];
  int d = dst[wid];
  float sc = invO[s];
  const float4* xs = (const float4*)(x + (size_t)s * HDIM);
  float4 v = xs[lane];
  float* md = m + (size_t)d * HDIM + lane * 4;
  atomicAdd(md + 0, v.x * sc);
  atomicAdd(md + 1, v.y * sc);
  atomicAdd(md + 2, v.z * sc);
  atomicAdd(md + 3, v.w * sc);
}

// ---------------- finalize (in place): x = m * inv_in[n] + bias[t] ; optional relu ----------------
__global__ void gcn_finalize(float* __restrict__ xm, const float* __restrict__ invI,
                             const float* __restrict__ bias, int total, int do_relu) {
  int i = blockIdx.x * blockDim.x + threadIdx.x;
  if (i < total) {
    int node = i >> 7;
    int t = i & (HDIM - 1);
    float v = xm[i] * invI[node] + bias[t];
    xm[i] = (do_relu && v < 0.0f) ? 0.0f : v;
  }
}

// ---------------- center pooling: xp = x[p0] * x[p1] ----------------
__global__ void gcn_pair_pool(const int* __restrict__ pair, const float* __restrict__ x,
                              float* __restrict__ xp, int Bn) {
  int i = blockIdx.x * blockDim.x + threadIdx.x;
  if (i < Bn * HDIM) {
    int b = i >> 7;
    int t = i & (HDIM - 1);
    int p0 = pair[b];
    int p1 = pair[Bn + b];
    xp[i] = x[(size_t)p0 * HDIM + t] * x[(size_t)p1 * HDIM + t];
  }
}

// ---------------- MLP layer 1 via WMMA f32 16x16x4: h = relu(xp @ W1 + b1) ----------------
// grid: (M/16, H/16), block: 32 (one wave per 16x16 output tile)
__global__ void gcn_mlp1_wmma(const float* __restrict__ xp, const float* __restrict__ W1,
                              const float* __restrict__ b1, float* __restrict__ h) {
  int tm = blockIdx.x;            // M tile
  int tn = blockIdx.y;            // N tile
  int lane = threadIdx.x;         // wave32
  int hh = lane >> 4;             // lane half (0/1)
  int idx = lane & 15;            // row (A) / col (B,D) within tile

  const float* Arow = xp + (size_t)(tm * 16 + idx) * HDIM;  // A: row M=idx
  const float* Bcol = W1 + tn * 16 + idx;                   // B: col N=idx, row stride HDIM

  v8f acc = {};
#pragma unroll
  for (int ks = 0; ks < HDIM / 4; ++ks) {
    int k0 = ks * 4 + hh * 2;     // lane-half K offset per ISA layout
    v2f afrag, bfrag;
    afrag[0] = Arow[k0];
    afrag[1] = Arow[k0 + 1];
    bfrag[0] = Bcol[(size_t)k0 * HDIM];
    bfrag[1] = Bcol[(size_t)(k0 + 1) * HDIM];
    // (neg_a, A, neg_b, B, c_mod, C, reuse_a, reuse_b)
    acc = __builtin_amdgcn_wmma_f32_16x16x4_f32(false, afrag, false, bfrag,
                                                (short)0, acc, false, false);
  }

  // D layout: VGPR j holds D[j + hh*8][idx]
#pragma unroll
  for (int j = 0; j < 8; ++j) {
    int m = tm * 16 + hh * 8 + j;
    int n = tn * 16 + idx;
    float v = acc[j] + b1[n];
    h[(size_t)m * HDIM + n] = v > 0.0f ? v : 0.0f;
  }
}

// ---------------- MLP layer 2: out = h @ W2 + b2 (dot per row, one wave per row) ----------------
__global__ void gcn_mlp2(const float* __restrict__ h, const float* __restrict__ W2,
                         const float* __restrict__ b2, float* __restrict__ out, int Bn) {
  int wid = (blockIdx.x * blockDim.x + threadIdx.x) >> 5;
  int lane = threadIdx.x & 31;
  if (wid >= Bn) return;
  const float* hr = h + (size_t)wid * HDIM;
  float s = 0.0f;
#pragma unroll
  for (int t = 0; t < HDIM; t += 32) s += hr[t + lane] * W2[t + lane];
#pragma unroll
  for (int off = 16; off > 0; off >>= 1) s += __shfl_xor(s, off, 32);
  if (lane == 0) out[wid] = s + b2[0];
}

extern "C" void kernel_launch(void* const* d_in, const int* in_sizes, int n_in,
                              void* d_out, int out_size, void* d_ws, size_t ws_size,
                              hipStream_t stream) {
  const int*   z       = (const int*)d_in[0];
  const int*   src     = (const int*)d_in[1];
  const int*   dst     = (const int*)d_in[2];
  const int*   pair    = (const int*)d_in[3];
  const float* z_table = (const float*)d_in[4];
  const float* bias0   = (const float*)d_in[5];
  const float* bias1   = (const float*)d_in[6];
  const float* bias2   = (const float*)d_in[7];
  const float* W1      = (const float*)d_in[8];
  const float* b1      = (const float*)d_in[9];
  const float* W2      = (const float*)d_in[10];
  const float* b2      = (const float*)d_in[11];

  const int N  = in_sizes[0];
  const int E  = in_sizes[1];
  const int Bn = in_sizes[3] / 2;
  const int NH = N * HDIM;

  // workspace layout (floats)
  float* ws   = (float*)d_ws;
  float* degO = ws;                      // N  -> becomes inv_out
  float* degI = degO + N;                // N  -> becomes inv_in
  float* xA   = degI + N;                // N*H
  float* xB   = xA + (size_t)NH;         // N*H
  float* xp   = xB + (size_t)NH;         // Bn*H
  float* hbuf = xp + (size_t)Bn * HDIM;  // Bn*H

  const int T = 256;

  // 1) degrees -> inv sqrt (degO/degI are contiguous: single zero fill of 2N)
  gcn_fill_zero<<<(2 * N + T - 1) / T, T, 0, stream>>>(degO, 2 * N);
  gcn_degrees<<<(E + T - 1) / T, T, 0, stream>>>(src, dst, degO, degI, E);
  gcn_inv_sqrt<<<(N + T - 1) / T, T, 0, stream>>>(degO, degI, N);

  // 2) embedding gather into xA
  gcn_gather_embed<<<(NH + T - 1) / T, T, 0, stream>>>(z, z_table, xA, NH);

  // 3) three conv layers, ping-pong xA/xB (accumulator finalized in place)
  const float* biases[3] = {bias0, bias1, bias2};
  float* xcur = xA;
  float* xnxt = xB;
  const int edgeBlocks = (E * 32 + T - 1) / T;  // one wave per edge
  for (int layer = 0; layer < 3; ++layer) {
    gcn_fill_zero<<<(NH + T - 1) / T, T, 0, stream>>>(xnxt, NH);
    gcn_edge_scatter<<<edgeBlocks, T, 0, stream>>>(src, dst, xcur, degO, xnxt, E);
    gcn_finalize<<<(NH + T - 1) / T, T, 0, stream>>>(xnxt, degI, biases[layer], NH,
                                                     layer < 2 ? 1 : 0);
    float* tmp = xcur; xcur = xnxt; xnxt = tmp;
  }

  // 4) center pooling
  gcn_pair_pool<<<(Bn * HDIM + T - 1) / T, T, 0, stream>>>(pair, xcur, xp, Bn);

  // 5) MLP layer 1 via WMMA (f32 16x16x4), one wave per 16x16 tile
  dim3 g1(Bn / 16, HDIM / 16);
  gcn_mlp1_wmma<<<g1, 32, 0, stream>>>(xp, W1, b1, hbuf);

  // 6) MLP layer 2 (dot per row)
  gcn_mlp2<<<(Bn * 32 + T - 1) / T, T, 0, stream>>>(hbuf, W2, b2, (float*)d_out, Bn);
}